// GATEncoder_26448408609248
// MI455X (gfx1250) — compile-verified
//
#include <hip/hip_runtime.h>

// ---------------------------------------------------------------------------
// GAT encoder for MI455X (gfx1250, wave32). Dense GEMMs use the CDNA5 native
// fp32 matrix op V_WMMA_F32_16X16X4_F32 (numerics match the fp32 reference).
// W panels are staged in LDS pair-interleaved over k so each B fragment is a
// single aligned ds_load_b64 straight into a WMMA-legal even VGPR pair. Each
// wave computes a 16x(CB*16) strip with CB accumulators reusing one A
// fragment (b64 load) per k-step. Edge softmax/aggregation is edge-parallel
// with float atomics (segment-max via order-preserving int mapping).
// ---------------------------------------------------------------------------

typedef __attribute__((ext_vector_type(2))) float v2f;
typedef __attribute__((ext_vector_type(8))) float v8f;

#define TPB 256
#define WAVES_PER_BLOCK 8

__device__ __forceinline__ unsigned fmap(float x) {
  unsigned u = __float_as_uint(x);
  return (u & 0x80000000u) ? ~u : (u | 0x80000000u);
}
__device__ __forceinline__ float funmap(unsigned u) {
  return __uint_as_float((u & 0x80000000u) ? (u & 0x7fffffffu) : ~u);
}

__global__ void fill_f32(float* __restrict__ p, float v, int count) {
  int t = blockIdx.x * blockDim.x + threadIdx.x;
  if (t < count) p[t] = v;
}
__global__ void fill_u32(unsigned* __restrict__ p, unsigned v, int count) {
  int t = blockIdx.x * blockDim.x + threadIdx.x;
  if (t < count) p[t] = v;
}

// ---------------------------------------------------------------------------
// C = X[n x K] @ W[K x ncols] with V_WMMA_F32_16X16X4_F32.
//   Block: 8 waves -> 128 rows x (CB*16) cols. Grid: (row groups, col strips).
//   LDS: KB x SW panel of W, pair-interleaved over k:
//        Wt[((k>>1)*SW + c)*2 + (k&1)] = W[k0+k][colBase+c]
//   so B frag {W[ka][c], W[ka+1][c]} = one contiguous aligned b64 (ka even).
// A frag (16x4 f32): lanes 0-15 rows K,K+1; lanes 16-31 rows K+2,K+3.
// C/D  (16x16 f32): VGPR i -> row i (lanes 0-15) / row i+8 (lanes 16-31).
// K must be a multiple of KB (128/256 here).
// ---------------------------------------------------------------------------
template <int CB, int KB>
__global__ __launch_bounds__(TPB) void wmma_gemm_f32_lds(
    const float* __restrict__ X, const float* __restrict__ W,
    float* __restrict__ Hout, int n, int K, int ncols) {
  constexpr int SW = CB * 16;  // strip width
  __shared__ __align__(16) float Wt[KB * SW];

  const int lane = threadIdx.x & 31;
  const int wave = threadIdx.x >> 5;
  const int tilesM = (n + 15) >> 4;
  const int tm = blockIdx.x * WAVES_PER_BLOCK + wave;
  const int tmc = (tm < tilesM) ? tm : (tilesM - 1);
  const int colBase = blockIdx.y * SW;
  const int khalf = (lane >> 4) << 1;  // 0 | 2
  const int rlane = lane & 15;

  // Clamp load row so every lane issues a valid, branch-free address.
  int row = tmc * 16 + rlane;
  if (row >= n) row = n - 1;
  const float* Xrow = X + (size_t)row * K;
  const v2f* Wt2 = (const v2f*)Wt;

  v8f acc[CB];
#pragma unroll
  for (int j = 0; j < CB; ++j) acc[j] = (v8f){};

  for (int k0 = 0; k0 < K; k0 += KB) {
    __syncthreads();
    // Cooperative load of W[k0..k0+KB) x [colBase..colBase+SW) into LDS,
    // pair-interleaved over k so B fragments are contiguous b64s.
#pragma unroll
    for (int idx = threadIdx.x; idx < KB * SW; idx += TPB) {
      int kk = idx / SW;
      int cc = idx - kk * SW;
      Wt[((((kk >> 1) * SW) + cc) << 1) | (kk & 1)] =
          W[(size_t)(k0 + kk) * ncols + colBase + cc];
    }
    // Prefetch next chunks while this one is consumed.
    if (k0 + KB < K) {
      __builtin_prefetch(Xrow + k0 + KB, 0, 0);
      __builtin_prefetch(W + (size_t)(k0 + KB) * ncols + colBase, 0, 0);
    }
    __syncthreads();

#pragma unroll
    for (int kk = 0; kk < KB; kk += 4) {
      const int ka = kk + khalf;  // even
      v2f a = *(const v2f*)(Xrow + k0 + ka);  // one b64 load, reused CB times
#pragma unroll
      for (int j = 0; j < CB; ++j) {
        v2f b = Wt2[(ka >> 1) * SW + j * 16 + rlane];  // one ds_load_b64
        acc[j] = __builtin_amdgcn_wmma_f32_16x16x4_f32(
            /*neg_a=*/false, a, /*neg_b=*/false, b,
            /*c_mod=*/(short)0, acc[j], /*reuse_a=*/false, /*reuse_b=*/false);
      }
    }
  }

  if (tm < tilesM) {
    const int rbase = tm * 16 + ((lane >> 4) << 3);
    if (rbase + 7 < n) {
      // Full tile: unguarded stores (N % 16 == 0 path).
#pragma unroll
      for (int j = 0; j < CB; ++j) {
        const int c = colBase + j * 16 + rlane;
#pragma unroll
        for (int i = 0; i < 8; ++i)
          Hout[(size_t)(rbase + i) * ncols + c] = acc[j][i];
      }
    } else {
#pragma unroll
      for (int j = 0; j < CB; ++j) {
        const int c = colBase + j * 16 + rlane;
#pragma unroll
        for (int i = 0; i < 8; ++i) {
          int r = rbase + i;
          if (r < n) Hout[(size_t)r * ncols + c] = acc[j][i];
        }
      }
    }
  }
}

// Per-node attention scores: s_src[n,h] = sum_c H[n,h*cc+c]*a_src[h,c], etc.
__global__ void scores_kernel(const float* __restrict__ H,
                              const float* __restrict__ a_src,
                              const float* __restrict__ a_dst,
                              float* __restrict__ s_src,
                              float* __restrict__ s_dst,
                              int n, int heads, int cc) {
  int t = blockIdx.x * blockDim.x + threadIdx.x;
  if (t >= n * heads) return;
  int node = t / heads;
  int h = t - node * heads;
  const float* hp = H + (size_t)node * heads * cc + h * cc;
  const float* as = a_src + h * cc;
  const float* ad = a_dst + h * cc;
  float ss = 0.0f, sd = 0.0f;
  for (int c = 0; c < cc; ++c) {
    float v = hp[c];
    ss += v * as[c];
    sd += v * ad[c];
  }
  s_src[t] = ss;
  s_dst[t] = sd;
}

__device__ __forceinline__ void edge_nodes(const long long* __restrict__ ei,
                                           int e, int E, int& s, int& d) {
  if (e < E) {
    s = (int)ei[e];
    d = (int)ei[E + e];
  } else {
    s = d = e - E;  // self loop
  }
}

// Pass 1: segment max of leaky-relu'd scores per (dst, head).
__global__ void edge_max_kernel(const long long* __restrict__ ei, int E, int n,
                                int heads, const float* __restrict__ s_src,
                                const float* __restrict__ s_dst,
                                unsigned* __restrict__ mInt) {
  int t = blockIdx.x * blockDim.x + threadIdx.x;
  int total = (E + n) * heads;
  if (t >= total) return;
  int e = t / heads;
  int h = t - e * heads;
  int s, d;
  edge_nodes(ei, e, E, s, d);
  float x = s_src[s * heads + h] + s_dst[d * heads + h];
  x = (x > 0.0f) ? x : 0.2f * x;
  atomicMax(&mInt[d * heads + h], fmap(x));
}

// Pass 2: denom[d,h] += exp(e - m[d,h])
__global__ void edge_den_kernel(const long long* __restrict__ ei, int E, int n,
                                int heads, const float* __restrict__ s_src,
                                const float* __restrict__ s_dst,
                                const unsigned* __restrict__ mInt,
                                float* __restrict__ denom) {
  int t = blockIdx.x * blockDim.x + threadIdx.x;
  int total = (E + n) * heads;
  if (t >= total) return;
  int e = t / heads;
  int h = t - e * heads;
  int s, d;
  edge_nodes(ei, e, E, s, d);
  float x = s_src[s * heads + h] + s_dst[d * heads + h];
  x = (x > 0.0f) ? x : 0.2f * x;
  float m = funmap(mInt[d * heads + h]);
  atomicAdd(&denom[d * heads + h], expf(x - m));
}

// Pass 3: agg[d, :] += alpha * H[s, :]. One thread per (edge, 8 channels).
__global__ void edge_agg_kernel(const long long* __restrict__ ei, int E, int n,
                                int heads, int cc,
                                const float* __restrict__ Hfeat,
                                const float* __restrict__ s_src,
                                const float* __restrict__ s_dst,
                                const unsigned* __restrict__ mInt,
                                const float* __restrict__ denom,
                                float* __restrict__ agg) {
  int hc = heads * cc;
  int lanesPerEdge = hc >> 3;  // 8 channels per thread
  long long t = (long long)blockIdx.x * blockDim.x + threadIdx.x;
  long long total = (long long)(E + n) * lanesPerEdge;
  if (t >= total) return;
  int e = (int)(t / lanesPerEdge);
  int part = (int)(t - (long long)e * lanesPerEdge);
  int ch0 = part << 3;
  int h = ch0 / cc;
  int s, d;
  edge_nodes(ei, e, E, s, d);
  float x = s_src[s * heads + h] + s_dst[d * heads + h];
  x = (x > 0.0f) ? x : 0.2f * x;
  float m = funmap(mInt[d * heads + h]);
  float den = denom[d * heads + h];
  float alpha = expf(x - m) / den;
  const float* hs = Hfeat + (size_t)s * hc + ch0;
  float* ap = agg + (size_t)d * hc + ch0;
#pragma unroll
  for (int i = 0; i < 8; ++i) atomicAdd(&ap[i], hs[i] * alpha);
}

// out[n,:] = (ELU?)(LayerNorm(agg[n,:] + bias) * g + b). One wave per node.
__global__ void ln_act_kernel(const float* __restrict__ agg,
                              const float* __restrict__ bias,
                              const float* __restrict__ g,
                              const float* __restrict__ b,
                              float* __restrict__ out,
                              int n, int hc, int doElu) {
  int wid = (blockIdx.x * blockDim.x + threadIdx.x) >> 5;
  int lane = threadIdx.x & 31;
  if (wid >= n) return;
  int per = hc >> 5;  // 8 for hc=256, 1 for hc=32
  float vals[8];
  float sum = 0.0f;
  for (int i = 0; i < per; ++i) {
    int ch = lane * per + i;
    float v = agg[(size_t)wid * hc + ch] + bias[ch];
    vals[i] = v;
    sum += v;
  }
  for (int o = 16; o > 0; o >>= 1) sum += __shfl_xor(sum, o, 32);
  float mu = sum / (float)hc;
  float var = 0.0f;
  for (int i = 0; i < per; ++i) {
    float dd = vals[i] - mu;
    var += dd * dd;
  }
  for (int o = 16; o > 0; o >>= 1) var += __shfl_xor(var, o, 32);
  var /= (float)hc;
  float inv = rsqrtf(var + 1e-5f);
  for (int i = 0; i < per; ++i) {
    int ch = lane * per + i;
    float y = (vals[i] - mu) * inv * g[ch] + b[ch];
    if (doElu) y = (y > 0.0f) ? y : (expf(y) - 1.0f);
    out[(size_t)wid * hc + ch] = y;
  }
}

__global__ void pool_sum_kernel(const float* __restrict__ h3,
                                const long long* __restrict__ batch,
                                float* __restrict__ pool,
                                float* __restrict__ cnt, int n, int outc) {
  int t = blockIdx.x * blockDim.x + threadIdx.x;
  if (t >= n * outc) return;
  int node = t / outc;
  int c = t - node * outc;
  int gid = (int)batch[node];
  atomicAdd(&pool[gid * outc + c], h3[(size_t)node * outc + c]);
  if (c == 0) atomicAdd(&cnt[gid], 1.0f);
}

__global__ void pool_div_kernel(const float* __restrict__ pool,
                                const float* __restrict__ cnt,
                                float* __restrict__ out, int gN, int outc) {
  int t = blockIdx.x * blockDim.x + threadIdx.x;
  if (t >= gN * outc) return;
  out[t] = pool[t] / fmaxf(cnt[t / outc], 1.0f);
}

// ---------------------------------------------------------------------------
static inline int blocks_for(long long total) {
  return (int)((total + TPB - 1) / TPB);
}

static void run_gat_layer(const float* Xin, const float* W, const float* a_s,
                          const float* a_d, int K, int heads, int cc,
                          const long long* ei, int E, int N, float* bufH,
                          float* bufAgg, float* s_src, float* s_dst,
                          unsigned* mInt, float* denom, hipStream_t stream) {
  int hc = heads * cc;
  int Et = E + N;

  fill_u32<<<blocks_for(N * heads), TPB, 0, stream>>>(mInt, 0u, N * heads);
  fill_f32<<<blocks_for(N * heads), TPB, 0, stream>>>(denom, 0.0f, N * heads);
  fill_f32<<<blocks_for((long long)N * hc), TPB, 0, stream>>>(bufAgg, 0.0f,
                                                              N * hc);

  int tilesM = (N + 15) / 16;
  int rowGroups = (tilesM + WAVES_PER_BLOCK - 1) / WAVES_PER_BLOCK;
  if (hc % 64 == 0) {
    dim3 grid(rowGroups, hc / 64);
    wmma_gemm_f32_lds<4, 32><<<grid, TPB, 0, stream>>>(Xin, W, bufH, N, K, hc);
  } else {
    dim3 grid(rowGroups, hc / 32);
    wmma_gemm_f32_lds<2, 32><<<grid, TPB, 0, stream>>>(Xin, W, bufH, N, K, hc);
  }

  scores_kernel<<<blocks_for(N * heads), TPB, 0, stream>>>(
      bufH, a_s, a_d, s_src, s_dst, N, heads, cc);

  edge_max_kernel<<<blocks_for((long long)Et * heads), TPB, 0, stream>>>(
      ei, E, N, heads, s_src, s_dst, mInt);
  edge_den_kernel<<<blocks_for((long long)Et * heads), TPB, 0, stream>>>(
      ei, E, N, heads, s_src, s_dst, mInt, denom);
  edge_agg_kernel<<<blocks_for((long long)Et * (hc / 8)), TPB, 0, stream>>>(
      ei, E, N, heads, cc, bufH, s_src, s_dst, mInt, denom, bufAgg);
}

extern "C" void kernel_launch(void* const* d_in, const int* in_sizes, int n_in,
                              void* d_out, int out_size, void* d_ws,
                              size_t ws_size, hipStream_t stream) {
  const int F_IN = 128, H = 4, C = 64, HC = 256, OUTC = 32;
  const int N = in_sizes[0] / F_IN;
  const int E = in_sizes[1] / 2;
  const int G = out_size / OUTC;

  const float* x = (const float*)d_in[0];
  const long long* ei = (const long long*)d_in[1];
  const long long* batch = (const long long*)d_in[2];
  const float* W1 = (const float*)d_in[3];
  const float* as1 = (const float*)d_in[4];
  const float* ad1 = (const float*)d_in[5];
  const float* b1 = (const float*)d_in[6];
  const float* ln1g = (const float*)d_in[7];
  const float* ln1b = (const float*)d_in[8];
  const float* W2 = (const float*)d_in[9];
  const float* as2 = (const float*)d_in[10];
  const float* ad2 = (const float*)d_in[11];
  const float* b2 = (const float*)d_in[12];
  const float* ln2g = (const float*)d_in[13];
  const float* ln2b = (const float*)d_in[14];
  const float* W3 = (const float*)d_in[15];
  const float* as3 = (const float*)d_in[16];
  const float* ad3 = (const float*)d_in[17];
  const float* b3 = (const float*)d_in[18];
  const float* ln3g = (const float*)d_in[19];
  const float* ln3b = (const float*)d_in[20];

  // Workspace carve-up (all fp32-sized slots).
  float* bufH = (float*)d_ws;                   // N*HC
  float* bufAgg = bufH + (size_t)N * HC;        // N*HC
  float* bufX = bufAgg + (size_t)N * HC;        // N*HC (activated features)
  float* s_src = bufX + (size_t)N * HC;         // N*H
  float* s_dst = s_src + (size_t)N * H;         // N*H
  unsigned* mInt = (unsigned*)(s_dst + (size_t)N * H);  // N*H
  float* denom = (float*)(mInt + (size_t)N * H);        // N*H
  float* pool = denom + (size_t)N * H;          // G*OUTC
  float* cnt = pool + (size_t)G * OUTC;         // G
  (void)ws_size;
  (void)n_in;

  // ---- Layer 1: 128 -> 4x64 concat, LN, ELU ----
  run_gat_layer(x, W1, as1, ad1, F_IN, H, C, ei, E, N, bufH, bufAgg, s_src,
                s_dst, mInt, denom, stream);
  ln_act_kernel<<<blocks_for((long long)N * 32), TPB, 0, stream>>>(
      bufAgg, b1, ln1g, ln1b, bufX, N, HC, 1);

  // ---- Layer 2: 256 -> 4x64 concat, LN, ELU ----
  run_gat_layer(bufX, W2, as2, ad2, HC, H, C, ei, E, N, bufH, bufAgg, s_src,
                s_dst, mInt, denom, stream);
  ln_act_kernel<<<blocks_for((long long)N * 32), TPB, 0, stream>>>(
      bufAgg, b2, ln2g, ln2b, bufX, N, HC, 1);

  // ---- Layer 3: 256 -> 1x32 (mean over 1 head == identity), LN ----
  run_gat_layer(bufX, W3, as3, ad3, HC, 1, OUTC, ei, E, N, bufH, bufAgg, s_src,
                s_dst, mInt, denom, stream);
  ln_act_kernel<<<blocks_for((long long)N * 32), TPB, 0, stream>>>(
      bufAgg, b3, ln3g, ln3b, bufX, N, OUTC, 0);

  // ---- Global mean pool over graphs ----
  fill_f32<<<blocks_for(G * OUTC), TPB, 0, stream>>>(pool, 0.0f, G * OUTC);
  fill_f32<<<blocks_for(G), TPB, 0, stream>>>(cnt, 0.0f, G);
  pool_sum_kernel<<<blocks_for((long long)N * OUTC), TPB, 0, stream>>>(
      bufX, batch, pool, cnt, N, OUTC);
  pool_div_kernel<<<blocks_for(G * OUTC), TPB, 0, stream>>>(
      pool, cnt, (float*)d_out, G, OUTC);
}